// GCNModel_3332894622176
// MI455X (gfx1250) — compile-verified
//
#include <hip/hip_runtime.h>
#include <hip/hip_bf16.h>
#include <math.h>

typedef __attribute__((ext_vector_type(2))) float v2f;
typedef __attribute__((ext_vector_type(8))) float v8f;

#define F_IN  256
#define F_HID 50
#define F_OUT 40
#define LD1   64   // padded hidden width (layer-1 output / layer-2 K)
#define LD2   48   // padded layer-2 output width (3 x 16)

// ---------------------------------------------------------------------------
// Pack W1 -> Wp1 [64 x 256] column(K)-major zero-padded, b1 -> bp1[64],
//      W2 -> Wp2 [48 x 64]  column(K)-major zero-padded, b2 -> bp2[48].
// ---------------------------------------------------------------------------
__global__ void prep_kernel(const float* __restrict__ W1, const float* __restrict__ b1,
                            const float* __restrict__ W2, const float* __restrict__ b2,
                            float* __restrict__ Wp1, float* __restrict__ bp1,
                            float* __restrict__ Wp2, float* __restrict__ bp2) {
    const int stride = gridDim.x * blockDim.x;
    const int tid = blockIdx.x * blockDim.x + threadIdx.x;
    for (int i = tid; i < LD1 * F_IN; i += stride) {       // Wp1[n*256+k]
        int n = i >> 8, k = i & 255;
        Wp1[i] = (n < F_HID) ? W1[k * F_HID + n] : 0.0f;
    }
    for (int i = tid; i < LD1; i += stride)
        bp1[i] = (i < F_HID) ? b1[i] : 0.0f;
    for (int i = tid; i < LD2 * LD1; i += stride) {        // Wp2[n*64+k]
        int n = i >> 6, k = i & 63;
        Wp2[i] = (n < F_OUT && k < F_HID) ? W2[k * F_OUT + n] : 0.0f;
    }
    for (int i = tid; i < LD2; i += stride)
        bp2[i] = (i < F_OUT) ? b2[i] : 0.0f;
}

__global__ void zero_kernel(float* __restrict__ p, size_t n) {
    size_t i = (size_t)blockIdx.x * blockDim.x + threadIdx.x;
    if (i < n) p[i] = 0.0f;
}

// one thread per (edge or self-loop): deg[col] += 1
__global__ void degree_kernel(const int* __restrict__ ei, float* __restrict__ deg,
                              int E, int N) {
    int idx = blockIdx.x * blockDim.x + threadIdx.x;
    if (idx >= E + N) return;
    int col = (idx < E) ? ei[E + idx] : (idx - E);
    atomicAdd(&deg[col], 1.0f);
}

__global__ void dinv_kernel(float* __restrict__ deg, int N) {
    int i = blockIdx.x * blockDim.x + threadIdx.x;
    if (i >= N) return;
    float d = deg[i];
    deg[i] = (d > 0.0f) ? rsqrtf(d) : 0.0f;
}

// ---------------------------------------------------------------------------
// Fp32 WMMA GEMM:  Out[M x LDOUT] = A[M x K] * Bw^T + bias
//   A is row-major with lda == K; Bw is [n][k] (column-of-B major), ld == K.
//   One wave computes a 16x16 tile via V_WMMA_F32_16X16X4_F32, K stepped by 4.
//   A layout : lanes 0-15 rows M, v[0..1] = K{0,1}; lanes 16-31 = K{2,3}.
//   B layout : lanes 0-15 cols N, v[0..1] = K{0,1}; lanes 16-31 = K{2,3}.
//   C/D      : VGPR r -> (M = r + 8*(lane>>4), N = lane&15).
// ---------------------------------------------------------------------------
template<int K, int LDOUT>
__global__ void gemm_wmma(const float* __restrict__ A, const float* __restrict__ Bw,
                          const float* __restrict__ bias, float* __restrict__ Out) {
    const int lane = threadIdx.x & 31;
    const int wave = threadIdx.x >> 5;
    const int l15  = lane & 15;
    const int hi   = lane >> 4;          // 0: K+{0,1}, 1: K+{2,3}
    const int m0   = blockIdx.x * 16;
    const int n0   = wave * 16;

    float bv = bias[n0 + l15];           // bias depends only on N column
    v8f acc;
#pragma unroll
    for (int r = 0; r < 8; ++r) acc[r] = bv;

    const float* ap = A  + (size_t)(m0 + l15) * K + 2 * hi;
    const float* bp = Bw + (size_t)(n0 + l15) * K + 2 * hi;

#pragma unroll 4
    for (int k = 0; k < K; k += 4) {
        v2f a = *(const v2f*)(ap + k);
        v2f b = *(const v2f*)(bp + k);
        acc = __builtin_amdgcn_wmma_f32_16x16x4_f32(
            /*neg_a=*/false, a, /*neg_b=*/false, b,
            /*c_mod=*/(short)0, acc, /*reuse_a=*/false, /*reuse_b=*/false);
    }

    float* op = Out + (size_t)(m0 + 8 * hi) * LDOUT + n0 + l15;
#pragma unroll
    for (int r = 0; r < 8; ++r) op[(size_t)r * LDOUT] = acc[r];
}

// ---------------------------------------------------------------------------
// Edge aggregation: one wave per edge; lanes stride the feature dim.
//   out[col] += dinv[row]*dinv[col] * h[row]   (coalesced gather + f32 atomics,
//   all node-feature traffic is L2-resident: 20 MB << 192 MB L2)
// ---------------------------------------------------------------------------
__global__ void agg_kernel(const float* __restrict__ h, float* __restrict__ out,
                           const float* __restrict__ dinv, const int* __restrict__ ei,
                           int E, int N, int F, int ldin, int ldout) {
    int wid  = (blockIdx.x * blockDim.x + threadIdx.x) >> 5;
    int lane = threadIdx.x & 31;
    if (wid >= E + N) return;
    int row, col;
    if (wid < E) { row = ei[wid]; col = ei[E + wid]; }
    else         { row = col = wid - E; }
    float norm = dinv[row] * dinv[col];
    const float* hp = h + (size_t)row * ldin;
    float* op = out + (size_t)col * ldout;
    for (int f = lane; f < F; f += 32)
        atomicAdd(op + f, norm * hp[f]);
}

__global__ void relu_kernel(float* __restrict__ p, size_t n) {
    size_t i = (size_t)blockIdx.x * blockDim.x + threadIdx.x;
    if (i < n) p[i] = fmaxf(p[i], 0.0f);
}

// one wave per node row of 40: log_softmax via shfl_xor reductions
__global__ void logsoftmax_kernel(const float* __restrict__ g, float* __restrict__ out, int N) {
    int wid  = (blockIdx.x * blockDim.x + threadIdx.x) >> 5;
    int lane = threadIdx.x & 31;
    if (wid >= N) return;
    const float* p = g + (size_t)wid * F_OUT;
    float v0 = p[lane];
    float v1 = (lane < F_OUT - 32) ? p[32 + lane] : -INFINITY;
    float m = fmaxf(v0, v1);
#pragma unroll
    for (int off = 16; off > 0; off >>= 1) m = fmaxf(m, __shfl_xor(m, off, 32));
    float s = expf(v0 - m) + ((lane < F_OUT - 32) ? expf(v1 - m) : 0.0f);
#pragma unroll
    for (int off = 16; off > 0; off >>= 1) s += __shfl_xor(s, off, 32);
    float lse = m + logf(s);
    out[(size_t)wid * F_OUT + lane] = v0 - lse;
    if (lane < F_OUT - 32) out[(size_t)wid * F_OUT + 32 + lane] = v1 - lse;
}

// ---------------------------------------------------------------------------
extern "C" void kernel_launch(void* const* d_in, const int* in_sizes, int n_in,
                              void* d_out, int out_size, void* d_ws, size_t ws_size,
                              hipStream_t stream) {
    const float* x  = (const float*)d_in[0];
    const int*   ei = (const int*)  d_in[1];   // [2,E] flat int32
    const float* W1 = (const float*)d_in[2];
    const float* b1 = (const float*)d_in[3];
    const float* W2 = (const float*)d_in[4];
    const float* b2 = (const float*)d_in[5];
    float* out = (float*)d_out;

    const int N = in_sizes[0] / F_IN;     // 100000 (divisible by 16)
    const int E = in_sizes[1] / 2;        // 3200000
    const int TOT = E + N;                // edges incl. self-loops

    // workspace layout (floats): ~68 MB total
    float* ws   = (float*)d_ws;
    float* Wp1  = ws;                               // 64*256
    float* bp1  = Wp1 + LD1 * F_IN;                 // 64
    float* Wp2  = bp1 + LD1;                        // 48*64
    float* bp2  = Wp2 + LD2 * LD1;                  // 48
    float* dinv = bp2 + LD2;                        // N (deg then rsqrt in place)
    float* g1   = dinv + N;                         // N*64 (agg1 out, zero-padded cols)
    float* g2   = g1 + (size_t)N * LD1;             // N*40 (agg2 accumulator)
    float* hbuf = g2 + (size_t)N * F_OUT;           // N*64 (h1, reused for h2 ld=48)

    // 1) pack/pad weights
    prep_kernel<<<64, 256, 0, stream>>>(W1, b1, W2, b2, Wp1, bp1, Wp2, bp2);

    // 2) zero dinv | g1 | g2 (contiguous): N*(1+64+40) floats
    size_t zn = (size_t)N * (1 + LD1 + F_OUT);
    zero_kernel<<<(unsigned)((zn + 255) / 256), 256, 0, stream>>>(dinv, zn);

    // 3) degrees -> dinv
    degree_kernel<<<(TOT + 255) / 256, 256, 0, stream>>>(ei, dinv, E, N);
    dinv_kernel<<<(N + 255) / 256, 256, 0, stream>>>(dinv, N);

    // 4) h1 = x @ W1 + b1  (fp32 WMMA, 4 waves cover n-tiles 0..63)
    gemm_wmma<F_IN, LD1><<<N / 16, 128, 0, stream>>>(x, Wp1, bp1, hbuf);

    // 5) g1 = normalized scatter-add of h1 ; then ReLU (padded cols stay 0)
    agg_kernel<<<(TOT + 7) / 8, 256, 0, stream>>>(hbuf, g1, dinv, ei, E, N, F_HID, LD1, LD1);
    size_t rn = (size_t)N * LD1;
    relu_kernel<<<(unsigned)((rn + 255) / 256), 256, 0, stream>>>(g1, rn);

    // 6) h2 = g1 @ W2 + b2  (K padded to 64 with true zeros; 3 waves cover 48 cols)
    gemm_wmma<LD1, LD2><<<N / 16, 96, 0, stream>>>(g1, Wp2, bp2, hbuf);

    // 7) g2 = normalized scatter-add of h2
    agg_kernel<<<(TOT + 7) / 8, 256, 0, stream>>>(hbuf, g2, dinv, ei, E, N, F_OUT, LD2, F_OUT);

    // 8) log_softmax rows -> d_out
    logsoftmax_kernel<<<(N + 7) / 8, 256, 0, stream>>>(g2, out, N);
}